// Capsule_1855425872067
// MI455X (gfx1250) — compile-verified
//
#include <hip/hip_runtime.h>

typedef __attribute__((ext_vector_type(2))) float v2f;
typedef __attribute__((ext_vector_type(8))) float v8f;
typedef __attribute__((ext_vector_type(4))) int   v4i;

#define BATCH    64
#define NIN      2048
#define DIN      64
#define NCAP     32
#define DCAP     16
#define MOUT     (NCAP*DCAP)     // 512
#define CHUNK    64
#define NCHUNKS  (NIN/CHUNK)     // 32
#define THREADS  512

#if defined(__gfx1250__) && \
    __has_builtin(__builtin_amdgcn_global_load_async_to_lds_b128) && \
    __has_builtin(__builtin_amdgcn_s_wait_asynccnt)
#define USE_ASYNC 1
#else
#define USE_ASYNC 0
#endif

#if USE_ASYNC
typedef __attribute__((address_space(1))) v4i* gv4i_p;   // global (prints as __device__)
typedef __attribute__((address_space(3))) v4i* lv4i_p;   // LDS
__device__ __forceinline__ void async_cp16(const float* g, float* l) {
    // one B128 per lane, tracked on ASYNCcnt, lands directly in LDS
    __builtin_amdgcn_global_load_async_to_lds_b128((gv4i_p)g, (lv4i_p)l, 0, 0);
}
#endif

__global__ __launch_bounds__(THREADS, 1)
void capsule_route(const float* __restrict__ u_vecs,
                   const float* __restrict__ kern,
                   float* __restrict__ out)
{
    __shared__ float kern_lds[DIN][MOUT];     // 128 KB  (weights, staged once)
    __shared__ float uv_lds[2][CHUNK][DIN];   //  32 KB  (double-buffered u_vecs chunk)
    __shared__ float uh_lds[CHUNK][MOUT];     // 128 KB  (u_hat chunk, fp32)
    __shared__ float c_lds[CHUNK][NCAP];      //   8 KB  (coupling coeffs)
    __shared__ float v_lds[NCAP][DCAP];       //   2 KB  (l2-normalized o)

    const int b    = blockIdx.x;
    const int tid  = threadIdx.x;
    const int lane = tid & 31;
    const int wave = tid >> 5;

    // ---- stage 64x512 weight matrix into LDS once (reused 3*32 times) ----
    {
        const float4* src = (const float4*)kern;
        float4* dst = (float4*)(&kern_lds[0][0]);
        for (int i = tid; i < DIN*MOUT/4; i += THREADS) dst[i] = src[i];
    }
    __syncthreads();

    const int icap = tid >> 4;     // capsule owned in o-accumulation (0..31)
    const int jcap = tid & 15;     // dim owned in o-accumulation (0..15)

    for (int it = 0; it < 3; ++it) {
        float oacc = 0.0f;                       // o[b][icap][jcap] partial

#if USE_ASYNC
        {   // prefetch chunk 0 into buffer 0 (4096 floats, 2 x B128 per lane)
            const float* g = u_vecs + ((size_t)b * NIN) * DIN;
            float*       l = &uv_lds[0][0][0];
            async_cp16(g + 4*tid,        l + 4*tid);
            async_cp16(g + 2048 + 4*tid, l + 2048 + 4*tid);
        }
#endif

        for (int ch = 0; ch < NCHUNKS; ++ch) {
            const int buf = ch & 1;

#if USE_ASYNC
            // prefetch next chunk into the alternate buffer while waiting.
            // Async loads complete in order, so asynccnt<=2 releases the
            // older pair (this chunk); last chunk drains fully.
            if (ch + 1 < NCHUNKS) {
                const float* g = u_vecs + ((size_t)b*NIN + (size_t)(ch+1)*CHUNK)*DIN;
                float*       l = &uv_lds[buf ^ 1][0][0];
                async_cp16(g + 4*tid,        l + 4*tid);
                async_cp16(g + 2048 + 4*tid, l + 2048 + 4*tid);
                __builtin_amdgcn_s_wait_asynccnt(2);
            } else {
                __builtin_amdgcn_s_wait_asynccnt(0);
            }
#else
            {   // synchronous staging fallback
                const float4* src =
                    (const float4*)(u_vecs + ((size_t)b*NIN + (size_t)ch*CHUNK)*DIN);
                float4* dst = (float4*)(&uv_lds[buf][0][0]);
                for (int i = tid; i < CHUNK*DIN/4; i += THREADS) dst[i] = src[i];
            }
#endif
            __syncthreads();

            // ---- WMMA GEMM: (64x64) @ (64x512) -> uh_lds, exact fp32 ----
            // 4 M-tiles x 32 N-tiles (16x16); K=64 as 16 x V_WMMA_F32_16X16X4_F32
            const int l15 = lane & 15;
            const int khi = (lane >> 4) << 1;    // lanes 16-31 carry K+2 / K+3
            #pragma unroll
            for (int ntb = 0; ntb < 2; ++ntb) {
                const int nt   = wave*2 + ntb;   // 16 waves x 2 = 32 N-tiles
                const int ncol = nt*16 + l15;
                v2f bf[16];                       // B fragments hoisted over mt
                #pragma unroll
                for (int k = 0; k < 16; ++k) {
                    const int kb = k*4 + khi;
                    bf[k].x = kern_lds[kb][ncol];
                    bf[k].y = kern_lds[kb+1][ncol];
                }
                #pragma unroll
                for (int mt = 0; mt < 4; ++mt) {
                    const int mrow = mt*16 + l15;
                    v8f acc = {};
                    #pragma unroll
                    for (int k = 0; k < 16; ++k) {
                        const int kb = k*4 + khi;
                        v2f af;
                        af.x = uv_lds[buf][mrow][kb];
                        af.y = uv_lds[buf][mrow][kb+1];
                        acc = __builtin_amdgcn_wmma_f32_16x16x4_f32(
                                false, af, false, bf[k], (short)0, acc,
                                false, false);
                    }
                    // D layout: VGPR r -> row mt*16 + r (+8 for lanes 16-31)
                    const int rbase = mt*16 + ((lane >> 4) << 3);
                    #pragma unroll
                    for (int r = 0; r < 8; ++r)
                        uh_lds[rbase + r][ncol] = acc[r];
                }
            }
            __syncthreads();

            // ---- logits b = <v_prev, u_hat> and softmax over capsules ----
            // (iteration 0: b == 0 -> c == 1/32, skip entirely)
            if (it > 0) {
                const int nl  = tid >> 3;   // row in chunk (0..63)
                const int sub = tid & 7;    // capsules sub, sub+8, sub+16, sub+24
                float bv[4];
                float bmax = -3.0e38f;
                #pragma unroll
                for (int q = 0; q < 4; ++q) {
                    const int i = sub + q*8;
                    float s = 0.0f;
                    #pragma unroll
                    for (int j = 0; j < DCAP; ++j)
                        s += v_lds[i][j] * uh_lds[nl][i*DCAP + j];
                    bv[q] = s;
                    bmax = fmaxf(bmax, s);
                }
                // butterfly over the 8 lanes sharing this n (wave32-safe)
                bmax = fmaxf(bmax, __shfl_xor(bmax, 1, 32));
                bmax = fmaxf(bmax, __shfl_xor(bmax, 2, 32));
                bmax = fmaxf(bmax, __shfl_xor(bmax, 4, 32));
                float es = 0.0f;
                #pragma unroll
                for (int q = 0; q < 4; ++q) { bv[q] = __expf(bv[q] - bmax); es += bv[q]; }
                es += __shfl_xor(es, 1, 32);
                es += __shfl_xor(es, 2, 32);
                es += __shfl_xor(es, 4, 32);
                const float inv = 1.0f / es;
                #pragma unroll
                for (int q = 0; q < 4; ++q)
                    c_lds[nl][sub + q*8] = bv[q] * inv;
            }
            __syncthreads();

            // ---- o[i][j] += sum_n c[i][n] * u_hat[n][i*16+j] ----
            if (it == 0) {
                float s = 0.0f;
                #pragma unroll 8
                for (int n = 0; n < CHUNK; ++n)
                    s += uh_lds[n][icap*DCAP + jcap];
                oacc += s * (1.0f / 32.0f);
            } else {
                #pragma unroll 8
                for (int n = 0; n < CHUNK; ++n)
                    oacc += c_lds[n][icap] * uh_lds[n][icap*DCAP + jcap];
            }
            // no trailing barrier needed: the staging barrier of the next
            // chunk is a full rendezvous before uh_lds/c_lds are rewritten,
            // and the prefetch targets the buffer freed by chunk ch-1.
        }

        // ---- |o_i|^2 via 16-lane butterfly (lanes 0-15 / 16-31 groups) ----
        float ss = oacc * oacc;
        ss += __shfl_xor(ss, 1, 32);
        ss += __shfl_xor(ss, 2, 32);
        ss += __shfl_xor(ss, 4, 32);
        ss += __shfl_xor(ss, 8, 32);

        if (it < 2) {
            // l2_normalize(o) -> v_lds for the next iteration's logits
            const float inv = 1.0f / sqrtf(fmaxf(ss, 1e-12f));
            v_lds[icap][jcap] = oacc * inv;
            __syncthreads();
        } else {
            // squash and emit
            const float s = ss + 1e-7f;
            const float scale = sqrtf(s) / (0.5f + s);
            out[((size_t)b*NCAP + icap)*DCAP + jcap] = oacc * scale;
        }
    }
}

extern "C" void kernel_launch(void* const* d_in, const int* in_sizes, int n_in,
                              void* d_out, int out_size, void* d_ws, size_t ws_size,
                              hipStream_t stream) {
    const float* u_vecs = (const float*)d_in[0];   // (64, 2048, 64) f32
    const float* kern   = (const float*)d_in[1];   // (1, 64, 512)  f32
    float* out = (float*)d_out;                    // (64, 32, 16)  f32
    (void)in_sizes; (void)n_in; (void)out_size; (void)d_ws; (void)ws_size;
    capsule_route<<<BATCH, THREADS, 0, stream>>>(u_vecs, kern, out);
}